// HistogramLoss_64579128263084
// MI455X (gfx1250) — compile-verified
//
#include <hip/hip_runtime.h>

#define NB    256
#define BATCH 8
#define CHANS 24            // B*3
#define HPIX  262144        // 512*512
#define TOTAL 6291456       // 8*3*512*512
#define HBLK  64            // histogram blocks per channel
#define PPB   (HPIX / HBLK) // 4096 pixels per histogram block
#define APPLY_BLOCKS (TOTAL / (256 * 4)) // 6144

typedef __attribute__((ext_vector_type(2))) float v2f;
typedef __attribute__((ext_vector_type(8))) float v8f;

// ---------------------------------------------------------------------------
// quantize: clip(rint(((x+1)*0.5*255) * m), 0, 255)  (rintf == round-half-even
// == jnp.round; (x+1)*127.5 rounds identically to ((x+1)*0.5)*255)
__device__ __forceinline__ int quantize(float x, float m) {
    float q = rintf((x + 1.0f) * 127.5f * m);
    q = fminf(fmaxf(q, 0.0f), 255.0f);
    return (int)q;
}

// ---------------------------------------------------------------------------
// Kernel 0: zero the global histograms.
__global__ __launch_bounds__(256) void k_init(int* hist_s, int* hist_r) {
    int i = blockIdx.x * 256 + threadIdx.x;
    if (i < CHANS * NB) { hist_s[i] = 0; hist_r[i] = 0; }
}

// ---------------------------------------------------------------------------
// Kernel 1: per-channel 256-bin histograms of quantized ref*mask_src (source)
// and target*mask_tar (reference). LDS histograms (ds_add), integer global
// atomics at the end (order-independent -> deterministic).
__global__ __launch_bounds__(256) void k_hist(const float* __restrict__ ref,
                                              const float* __restrict__ tar,
                                              const float* __restrict__ msrc,
                                              const float* __restrict__ mtar,
                                              int* __restrict__ hist_s,
                                              int* __restrict__ hist_r) {
    __shared__ int hs[NB];
    __shared__ int hr[NB];
    const int t = threadIdx.x;
    hs[t] = 0; hr[t] = 0;
    __syncthreads();

    const int ch   = blockIdx.x / HBLK;
    const int sub  = blockIdx.x % HBLK;
    const int b    = ch / 3;
    const size_t pix0 = (size_t)sub * PPB;

    const float4* R  = (const float4*)(ref  + (size_t)ch * HPIX + pix0);
    const float4* T  = (const float4*)(tar  + (size_t)ch * HPIX + pix0);
    const float4* MS = (const float4*)(msrc + (size_t)b  * HPIX + pix0);
    const float4* MT = (const float4*)(mtar + (size_t)b  * HPIX + pix0);

    #pragma unroll
    for (int it = 0; it < PPB / (256 * 4); ++it) {
        const int k = it * 256 + t;
        const float4 r4  = R[k];
        const float4 t4  = T[k];
        const float4 ms4 = MS[k];
        const float4 mt4 = MT[k];
        atomicAdd(&hs[quantize(r4.x, ms4.x)], 1);
        atomicAdd(&hs[quantize(r4.y, ms4.y)], 1);
        atomicAdd(&hs[quantize(r4.z, ms4.z)], 1);
        atomicAdd(&hs[quantize(r4.w, ms4.w)], 1);
        atomicAdd(&hr[quantize(t4.x, mt4.x)], 1);
        atomicAdd(&hr[quantize(t4.y, mt4.y)], 1);
        atomicAdd(&hr[quantize(t4.z, mt4.z)], 1);
        atomicAdd(&hr[quantize(t4.w, mt4.w)], 1);
    }
    __syncthreads();
    atomicAdd(&hist_s[ch * NB + t], hs[t]);
    atomicAdd(&hist_r[ch * NB + t], hr[t]);
}

// ---------------------------------------------------------------------------
// 256-element inclusive prefix sum on one wave32 via V_WMMA_F32_16X16X4_F32:
// P = Cnt(16x16) x U(16x16 upper-triangular ones), chained over 4 K-blocks,
// then cross-row offset fixup. Counts <= 2^18 are exact in f32.
// A layout (16x4 f32): lane L -> M = L&15, a[v] = A[M][2*(L>>4)+v]
// B layout (4x16 f32): lane L -> N = L&15, b[v] = B[2*(L>>4)+v][N]
// C/D layout:          lane L -> N = L&15, d[v] = D[v + 8*(L>>4)][N]
__device__ __forceinline__ void cumsum256_wmma(const float* in, float* out, int lane) {
    const int m  = lane & 15;
    const int kh = lane >> 4;
    v8f acc = {};
    #pragma unroll
    for (int kb = 0; kb < 4; ++kb) {
        const int k0 = kb * 4 + 2 * kh;
        v2f a, bb;
        a[0]  = in[m * 16 + k0];
        a[1]  = in[m * 16 + k0 + 1];
        bb[0] = (k0     <= m) ? 1.0f : 0.0f;  // U[k0][m]
        bb[1] = (k0 + 1 <= m) ? 1.0f : 0.0f;  // U[k0+1][m]
        acc = __builtin_amdgcn_wmma_f32_16x16x4_f32(
            /*neg_a=*/false, a, /*neg_b=*/false, bb,
            /*c_mod=*/(short)0, acc, /*reuse_a=*/false, /*reuse_b=*/false);
    }
    #pragma unroll
    for (int v = 0; v < 8; ++v) out[(v + 8 * kh) * 16 + m] = acc[v];
    __syncthreads();
    // Row offsets: O[M] = sum_{r<M} rowTotal[r], rowTotal[r] = P[r][15].
    float off[16];
    {
        float run = 0.0f;
        #pragma unroll
        for (int r = 0; r < 16; ++r) { off[r] = run; run += out[r * 16 + 15]; }
    }
    __syncthreads();
    #pragma unroll
    for (int v = 0; v < 8; ++v) {
        const int row = v + 8 * kh;
        out[row * 16 + m] += off[row];
    }
    __syncthreads();
}

// ---------------------------------------------------------------------------
// Kernel 2: build per-channel LUT (one wave per channel).
__global__ __launch_bounds__(32) void k_lut(const int* __restrict__ hist_s,
                                            const int* __restrict__ hist_r,
                                            float* __restrict__ lut) {
    __shared__ float cs[NB], cr[NB], cums[NB], cumr[NB];
    __shared__ float xp[NB], fpv[NB];
    __shared__ int sh_fs, sh_fr, sh_ns, sh_nr, sh_n;

    const int ch   = blockIdx.x;
    const int lane = threadIdx.x;
    const int* Hs = hist_s + ch * NB;
    const int* Hr = hist_r + ch * NB;

    for (int i = lane; i < NB; i += 32) {
        cs[i] = (float)Hs[i];
        cr[i] = (float)Hr[i];
    }
    __syncthreads();

    if (lane == 0) {
        int fs = 0, fr = 0, ns = 0, nr = 0, fsSet = 0, frSet = 0;
        for (int i = 0; i < NB; ++i) {
            if (Hs[i] > 0) { ns++; if (!fsSet) { fs = i; fsSet = 1; } }
            if (Hr[i] > 0) { nr++; if (!frSet) { fr = i; frSet = 1; } }
        }
        sh_fs = fs; sh_fr = fr; sh_ns = ns; sh_nr = nr;
        cs[fs] = 0.0f;   // drop smallest source value (maps to 0)
        cr[fr] = 0.0f;   // drop smallest reference value
    }
    __syncthreads();

    cumsum256_wmma(cs, cums, lane);
    cumsum256_wmma(cr, cumr, lane);

    if (lane == 0) {
        // Compact occupied reference bins (excluding first_r), ascending:
        // matches the stable argsort compaction in the reference.
        const float rden = fmaxf(cumr[NB - 1], 1.0f);
        const int fr = sh_fr;
        int n = 0;
        for (int bb = 0; bb < NB; ++bb) {
            if (Hr[bb] > 0 && bb != fr) {
                xp[n]  = cumr[bb] / rden;
                fpv[n] = (float)bb;
                ++n;
            }
        }
        sh_n = n;
    }
    __syncthreads();

    const int   n    = sh_n;
    const int   fs   = sh_fs;
    const bool  ok   = (sh_ns > 1) && (sh_nr > 1);
    const float sden = fmaxf(cums[NB - 1], 1.0f);
    float* L = lut + ch * NB;

    for (int i = lane; i < NB; i += 32) {
        float outv = 0.0f;
        if (ok && n > 0) {
            const float x = cums[i] / sden;
            // upper_bound: smallest k with xp[k] > x (xp strictly increasing)
            int lo = 0, hi = n;
            while (lo < hi) { int mid = (lo + hi) >> 1; if (xp[mid] > x) hi = mid; else lo = mid + 1; }
            const int j = lo - 1;
            float val;
            if (j < 0)           val = fpv[0];        // x <= xp[0]  -> left clamp
            else if (j >= n - 1) val = fpv[n - 1];    // x >= xp[-1] -> right clamp
            else {
                const float x0 = xp[j], x1 = xp[j + 1];
                val = fpv[j] + (x - x0) * (fpv[j + 1] - fpv[j]) / (x1 - x0);
            }
            outv = truncf(val);
        }
        if (i == fs) outv = 0.0f;
        if (!ok)     outv = 0.0f;
        L[i] = outv;
    }
}

// ---------------------------------------------------------------------------
// Kernel 3: matched = lut[q(ref*m)]; partial += |inp_img*m - matched*m|.
// Each block = 1024 consecutive elements of one channel; lut staged in LDS.
__global__ __launch_bounds__(256) void k_apply(const float* __restrict__ inp,
                                               const float* __restrict__ ref,
                                               const float* __restrict__ msrc,
                                               const float* __restrict__ lut,
                                               float* __restrict__ part) {
    __shared__ float sl[NB];
    __shared__ float red[256];
    const int t  = threadIdx.x;
    const int ch = blockIdx.x >> 8;              // 256 blocks per channel
    const int b  = ch / 3;
    sl[t] = lut[ch * NB + t];
    __syncthreads();

    const size_t pix0 = (size_t)(blockIdx.x & 255) * 1024 + (size_t)t * 4;
    const float4 in4 = *(const float4*)(inp  + (size_t)ch * HPIX + pix0);
    const float4 r4  = *(const float4*)(ref  + (size_t)ch * HPIX + pix0);
    const float4 m4  = *(const float4*)(msrc + (size_t)b  * HPIX + pix0);

    float s = 0.0f;
    {
        const float img = (in4.x + 1.0f) * 127.5f;
        s += fabsf(img * m4.x - sl[quantize(r4.x, m4.x)] * m4.x);
    }
    {
        const float img = (in4.y + 1.0f) * 127.5f;
        s += fabsf(img * m4.y - sl[quantize(r4.y, m4.y)] * m4.y);
    }
    {
        const float img = (in4.z + 1.0f) * 127.5f;
        s += fabsf(img * m4.z - sl[quantize(r4.z, m4.z)] * m4.z);
    }
    {
        const float img = (in4.w + 1.0f) * 127.5f;
        s += fabsf(img * m4.w - sl[quantize(r4.w, m4.w)] * m4.w);
    }

    red[t] = s;
    __syncthreads();
    #pragma unroll
    for (int off = 128; off > 0; off >>= 1) {
        if (t < off) red[t] += red[t + off];
        __syncthreads();
    }
    if (t == 0) part[blockIdx.x] = red[0];
}

// ---------------------------------------------------------------------------
// Kernel 4: deterministic final reduce -> mean.
__global__ __launch_bounds__(256) void k_reduce(const float* __restrict__ part,
                                                float* __restrict__ out) {
    __shared__ float red[256];
    const int t = threadIdx.x;
    float s = 0.0f;
    for (int k = t; k < APPLY_BLOCKS; k += 256) s += part[k];
    red[t] = s;
    __syncthreads();
    #pragma unroll
    for (int off = 128; off > 0; off >>= 1) {
        if (t < off) red[t] += red[t + off];
        __syncthreads();
    }
    if (t == 0) out[0] = red[0] * (1.0f / (float)TOTAL);
}

// ---------------------------------------------------------------------------
extern "C" void kernel_launch(void* const* d_in, const int* in_sizes, int n_in,
                              void* d_out, int out_size, void* d_ws, size_t ws_size,
                              hipStream_t stream) {
    const float* inp  = (const float*)d_in[0];  // input_data
    const float* tar  = (const float*)d_in[1];  // target_data
    const float* ref  = (const float*)d_in[2];  // ref_data
    const float* msrc = (const float*)d_in[3];  // mask_src
    const float* mtar = (const float*)d_in[4];  // mask_tar
    float* out = (float*)d_out;

    // Workspace layout
    int*   hist_s = (int*)d_ws;                        // 24*256 ints
    int*   hist_r = hist_s + CHANS * NB;               // 24*256 ints
    float* lut    = (float*)(hist_r + CHANS * NB);     // 24*256 floats
    float* part   = lut + CHANS * NB;                  // 6144 floats

    k_init<<<dim3(CHANS), dim3(256), 0, stream>>>(hist_s, hist_r);
    k_hist<<<dim3(CHANS * HBLK), dim3(256), 0, stream>>>(ref, tar, msrc, mtar,
                                                         hist_s, hist_r);
    k_lut<<<dim3(CHANS), dim3(32), 0, stream>>>(hist_s, hist_r, lut);
    k_apply<<<dim3(APPLY_BLOCKS), dim3(256), 0, stream>>>(inp, ref, msrc, lut, part);
    k_reduce<<<dim3(1), dim3(256), 0, stream>>>(part, out);
}